// GatingFeatureExtractor_31568009625764
// MI455X (gfx1250) — compile-verified
//
#include <hip/hip_runtime.h>
#include <math.h>

#define EPSF 1e-8f
#define NE 8
#define NC 1000
#define ROW_OUT 8029   // 8*1000 + 3*8 + 5

typedef __attribute__((ext_vector_type(2))) float v2f;
typedef __attribute__((ext_vector_type(4))) float v4f;
typedef __attribute__((ext_vector_type(8))) float v8f;

__global__ __launch_bounds__(256)
void gating_feat_kernel(const float* __restrict__ in, float* __restrict__ out) {
    __shared__ float pL[NE * NC];     // 32000 B
    __shared__ float lpL[NE * NC];    // 32000 B
    __shared__ float plogpS[NE];
    __shared__ int   amaxS[NE];
    __shared__ float crossPart[8];
    __shared__ float redME[8];
    __shared__ float redVS[8];

    const int b    = blockIdx.x;
    const int tid  = threadIdx.x;
    const int wave = tid >> 5;
    const int lane = tid & 31;
    const size_t inBase  = (size_t)b * (NE * NC);
    const size_t outBase = (size_t)b * ROW_OUT;

    // ---------------- Phase 1: wave w streams expert e = w (b128 NT loads) ----------------
    {
        const int e = wave;
        const v4f* pin4 = (const v4f*)(in + inBase + (size_t)e * NC);   // 16B aligned
        float*     pout = out + outBase + (size_t)e * NC;               // only 4B aligned (8029 odd)
        v4f*       pL4  = (v4f*)(pL  + e * NC);
        v4f*       lpL4 = (v4f*)(lpL + e * NC);

        float psum = 0.f;
        float m1 = -3.4e38f, m2 = -3.4e38f;
        int   amax = 0;

        for (int q = lane; q < NC / 4; q += 32) {          // 250 float4 chunks
            v4f pv = __builtin_nontemporal_load(&pin4[q]);
            const int c0 = q * 4;
            v4f lv;
#pragma unroll
            for (int j = 0; j < 4; ++j) lv[j] = __logf(pv[j] + EPSF);
            pL4[q]  = pv;
            lpL4[q] = lv;
#pragma unroll
            for (int j = 0; j < 4; ++j) {
                __builtin_nontemporal_store(pv[j], pout + c0 + j);   // identity copy
                psum = fmaf(pv[j], lv[j], psum);
                const float p = pv[j];
                if (p > m1)      { m2 = m1; m1 = p; amax = c0 + j; }
                else if (p > m2) { m2 = p; }
            }
        }
        // wave32 reduction: sum(psum), top-2(m1,m2), argmax (first index on ties)
        for (int off = 16; off >= 1; off >>= 1) {
            float om1 = __shfl_xor(m1, off, 32);
            float om2 = __shfl_xor(m2, off, 32);
            int   oam = __shfl_xor(amax, off, 32);
            float ops = __shfl_xor(psum, off, 32);
            psum += ops;
            if (om1 > m1 || (om1 == m1 && oam < amax)) {
                m2 = fmaxf(m1, om2); m1 = om1; amax = oam;
            } else {
                m2 = fmaxf(om1, m2);
            }
        }
        if (lane == 0) {
            out[outBase + 8000 + e] = -psum;     // expert_entropy
            out[outBase + 8008 + e] = m1;        // expert_confidence
            out[outBase + 8016 + e] = m1 - m2;   // expert_margin
            plogpS[e]    = psum;
            amaxS[e]     = amax;
            crossPart[e] = 0.f;
        }
    }
    __syncthreads();

    // ---------- Phase 2a: mixture entropy + unbiased variance (b128 LDS, 1 chunk/thread) ----------
    float mePart = 0.f, vsPart = 0.f;
    if (tid < NC / 4) {                       // threads 0..249, one float4-of-classes each
        const int q = tid;
        v4f pe[NE];
        v4f s = {0.f, 0.f, 0.f, 0.f};
#pragma unroll
        for (int e = 0; e < NE; ++e) { pe[e] = ((const v4f*)(pL + e * NC))[q]; s += pe[e]; }
        v4f mix = s * 0.125f;
#pragma unroll
        for (int j = 0; j < 4; ++j) mePart = fmaf(mix[j], __logf(mix[j] + EPSF), mePart);
#pragma unroll
        for (int e = 0; e < NE; ++e) {
            v4f d = pe[e] - mix;
#pragma unroll
            for (int j = 0; j < 4; ++j) vsPart = fmaf(d[j], d[j], vsPart);
        }
    }
    for (int off = 16; off >= 1; off >>= 1) {
        mePart += __shfl_xor(mePart, off, 32);
        vsPart += __shfl_xor(vsPart, off, 32);
    }
    if (lane == 0) { redME[wave] = mePart; redVS[wave] = vsPart; }

    // ---------- Phase 2b: cross[i,j] = sum_c p[i,c]*logp[j,c] via V_WMMA_F32_16X16X4_F32 ----------
    // Rows/cols 8..15 of the 16x16 tile duplicate experts 0..7 (row = lane&7): harmless garbage,
    // since C[i][j] depends only on A-row i / B-col j and we read only i,j < 8. No EXEC games.
    if (wave < 5) {                          // waves 0..4, 50 K-steps each over disjoint k-ranges
        v8f acc = {0.f, 0.f, 0.f, 0.f, 0.f, 0.f, 0.f, 0.f};
        const int row  = lane & 7;
        const int kadj = (lane < 16) ? 0 : 2;          // lanes 16..31 hold K+2, K+3
        const float* pRow  = pL  + row * NC;
        const float* lpRow = lpL + row * NC;
        const int k0 = wave * 200 + kadj;
        for (int s = 0; s < 50; ++s) {
            const int k = k0 + s * 4;                  // even index -> 8B-aligned b64 LDS loads
            v2f a  = *(const v2f*)(pRow  + k);
            v2f bb = *(const v2f*)(lpRow + k);
            acc = __builtin_amdgcn_wmma_f32_16x16x4_f32(
                /*neg_a=*/false, a, /*neg_b=*/false, bb,
                /*c_mod=*/(short)0, acc, /*reuse_a=*/false, /*reuse_b=*/false);
        }
        // C layout: VGPR v -> M=v (lanes 0..15), lane -> N. Need sum over i<j, i,j<8.
        float part = 0.f;
        if (lane < 8) {
#pragma unroll
            for (int v = 0; v < 8; ++v) if (v < lane) part += acc[v];
        }
        for (int off = 16; off >= 1; off >>= 1) part += __shfl_xor(part, off, 32);
        if (lane == 0) crossPart[wave] = part;
    }
    __syncthreads();

    // ---------------- Final scalars (thread 0) ----------------
    if (tid == 0) {
        float crossS = 0.f;
        for (int w = 0; w < 5; ++w) crossS += crossPart[w];
        float weighted = 0.f, plpSum = 0.f;
        for (int i = 0; i < NE; ++i) { weighted += plogpS[i] * (float)(7 - i); plpSum += plogpS[i]; }
        float me = 0.f, vs = 0.f;
        for (int w = 0; w < 8; ++w) { me += redME[w]; vs += redVS[w]; }
        const float mixtureEntropy = -me;
        int nu = 0;
        for (int i = 0; i < NE; ++i) {
            bool uniq = true;
            for (int j = 0; j < i; ++j) if (amaxS[j] == amaxS[i]) { uniq = false; break; }
            nu += uniq ? 1 : 0;
        }
        out[outBase + 8024] = (float)(nu - 1) * (1.0f / 7.0f);          // disagreement_ratio
        out[outBase + 8025] = (weighted - crossS) * (1.0f / 28.0f);      // mean_pairwise_kl
        out[outBase + 8026] = mixtureEntropy;                            // mixture_entropy
        out[outBase + 8027] = vs * (1.0f / (7.0f * (float)NC));          // post_var (ddof=1)
        out[outBase + 8028] = mixtureEntropy + plpSum * 0.125f;          // mutual_info
    }
}

extern "C" void kernel_launch(void* const* d_in, const int* in_sizes, int n_in,
                              void* d_out, int out_size, void* d_ws, size_t ws_size,
                              hipStream_t stream) {
    (void)n_in; (void)out_size; (void)d_ws; (void)ws_size;
    const float* p = (const float*)d_in[0];
    float* out = (float*)d_out;
    const int B = in_sizes[0] / (NE * NC);   // 8192
    gating_feat_kernel<<<B, 256, 0, stream>>>(p, out);
}